// DoubleLeakySNN_4870492914052
// MI455X (gfx1250) — compile-verified
//
#include <hip/hip_runtime.h>

typedef __attribute__((ext_vector_type(2))) float v2f;
typedef __attribute__((ext_vector_type(8))) float v8f;

#define B_TOT   16384
#define NI      512
#define NH      2048
#define TILES_N 2                   // 16-wide N tiles per wave
#define WAVE_N  (16 * TILES_N)      // 32 hidden columns per wave
#define WAVES_PER_BLOCK 8

// ---------------------------------------------------------------------------
// out[b] = b2[0]  (bias init; main kernel accumulates with atomics)
// ---------------------------------------------------------------------------
__global__ void snn_init_out(float* __restrict__ out, const float* __restrict__ b2) {
    int i = blockIdx.x * blockDim.x + threadIdx.x;
    if (i < B_TOT) out[i] = b2[0];
}

// ---------------------------------------------------------------------------
// Fused: cur = x@W1^T + b1 (f32 WMMA), 25-step LIF in registers,
//        out[b] += sum_h spk1[b,h]*W2[h] via shuffle reduce + atomicAdd.
// ---------------------------------------------------------------------------
__global__ __launch_bounds__(256) void snn_fused(
    const float* __restrict__ x,      // [B, NI]
    const float* __restrict__ W1,     // [NH, NI]
    const float* __restrict__ b1,     // [NH]
    const float* __restrict__ W2,     // [1, NH]
    const float* __restrict__ thr1,   // [NH]
    const float* __restrict__ thr2,   // [NH]
    const float* __restrict__ beta1p, // scalar
    const float* __restrict__ beta2p, // scalar
    const int*   __restrict__ stepsp, // scalar
    float* __restrict__ out)          // [B]
{
    const int lane  = threadIdx.x & 31;
    const int wave  = threadIdx.x >> 5;
    const int gwave = blockIdx.x * WAVES_PER_BLOCK + wave;

    const int n_groups = NH / WAVE_N;              // 64
    const int m_tile   = gwave / n_groups;         // 0..1023
    const int n_group  = gwave - m_tile * n_groups;

    const int half = lane >> 4;                    // 0 | 1
    const int ncol = lane & 15;                    // M row (A) / N col (B,C)

    // ---------------- GEMM1 via V_WMMA_F32_16X16X4_F32 ----------------
    v8f acc[TILES_N];
    #pragma unroll
    for (int t = 0; t < TILES_N; ++t) acc[t] = (v8f){};

    // A 16x4 f32 layout: lane = M, VGPR0/1 = K = 2*half + {0,1}
    const float* xrow  = x  + (size_t)(m_tile * 16 + ncol) * NI + 2 * half;
    // B 4x16 layout mirrored: lane = N, VGPR0/1 = K = 2*half + {0,1}
    const float* wrow0 = W1 + (size_t)(n_group * WAVE_N + ncol) * NI + 2 * half;

    #pragma unroll 4
    for (int k0 = 0; k0 < NI; k0 += 4) {
        v2f a;
        a.x = xrow[k0];
        a.y = xrow[k0 + 1];
        #pragma unroll
        for (int t = 0; t < TILES_N; ++t) {
            const float* wr = wrow0 + (size_t)(t * 16) * NI;
            v2f b;
            b.x = wr[k0];
            b.y = wr[k0 + 1];
            acc[t] = __builtin_amdgcn_wmma_f32_16x16x4_f32(
                false, a, false, b, (short)0, acc[t], false, false);
        }
    }

    // ---------------- LIF recurrence (register resident) ----------------
    const float bc1   = fminf(fmaxf(beta1p[0], 0.0f), 1.0f);
    const float bc2   = fminf(fmaxf(beta2p[0], 0.0f), 1.0f);
    const int   steps = stepsp[0];

    // C layout: VGPR r -> (M = r + 8*half, N = ncol)
    float s[TILES_N][8], m1[TILES_N][8], m2[TILES_N][8];
    float th1[TILES_N], th2[TILES_N], w2v[TILES_N];

    #pragma unroll
    for (int t = 0; t < TILES_N; ++t) {
        const int h = n_group * WAVE_N + t * 16 + ncol;
        const float bias = b1[h];
        th1[t] = thr1[h];
        th2[t] = thr2[h];
        w2v[t] = W2[h];
        #pragma unroll
        for (int r = 0; r < 8; ++r) {
            s[t][r]  = acc[t][r] + bias;   // initial carry spk1 = cur (analog)
            m1[t][r] = 0.0f;
            m2[t][r] = 0.0f;
        }
    }

    for (int it = 0; it < steps; ++it) {
        #pragma unroll
        for (int t = 0; t < TILES_N; ++t) {
            #pragma unroll
            for (int r = 0; r < 8; ++r) {
                // layer 1: reset from OLD mem, update, spike
                float rst1 = (m1[t][r] > th1[t]) ? th1[t] : 0.0f;
                m1[t][r]   = __builtin_fmaf(bc1, m1[t][r], s[t][r]) - rst1;
                float spk2 = (m1[t][r] > th1[t]) ? 1.0f : 0.0f;
                // layer 2
                float rst2 = (m2[t][r] > th2[t]) ? th2[t] : 0.0f;
                m2[t][r]   = __builtin_fmaf(bc2, m2[t][r], spk2) - rst2;
                s[t][r]    = (m2[t][r] > th2[t]) ? 1.0f : 0.0f;
            }
        }
    }

    // ---------------- out[b] += sum_h spk * W2[h] ----------------
    #pragma unroll
    for (int r = 0; r < 8; ++r) {
        float v = 0.0f;
        #pragma unroll
        for (int t = 0; t < TILES_N; ++t)
            v = __builtin_fmaf(s[t][r], w2v[t], v);
        // reduce over N: the 16 lanes of this half (masks stay within a half)
        v += __shfl_xor(v, 1);
        v += __shfl_xor(v, 2);
        v += __shfl_xor(v, 4);
        v += __shfl_xor(v, 8);
        if (ncol == 0)
            atomicAdd(&out[m_tile * 16 + half * 8 + r], v);
    }
}

// ---------------------------------------------------------------------------
extern "C" void kernel_launch(void* const* d_in, const int* in_sizes, int n_in,
                              void* d_out, int out_size, void* d_ws, size_t ws_size,
                              hipStream_t stream) {
    const float* x     = (const float*)d_in[0];
    const float* W1    = (const float*)d_in[1];
    const float* b1    = (const float*)d_in[2];
    const float* W2    = (const float*)d_in[3];
    const float* b2    = (const float*)d_in[4];
    const float* beta1 = (const float*)d_in[5];
    const float* beta2 = (const float*)d_in[6];
    const float* thr1  = (const float*)d_in[7];
    const float* thr2  = (const float*)d_in[8];
    const int*   steps = (const int*)d_in[9];
    float* out = (float*)d_out;

    snn_init_out<<<(B_TOT + 255) / 256, 256, 0, stream>>>(out, b2);

    const int total_waves = (B_TOT / 16) * (NH / WAVE_N);   // 1024 * 64
    const int blocks = total_waves / WAVES_PER_BLOCK;       // 8192
    snn_fused<<<blocks, 256, 0, stream>>>(x, W1, b1, W2, thr1, thr2,
                                          beta1, beta2, steps, out);
}